// Routing_9045201125830
// MI455X (gfx1250) — compile-verified
//
#include <hip/hip_runtime.h>
#include <hip/hip_bf16.h>

typedef __attribute__((ext_vector_type(2))) float v2f;
typedef __attribute__((ext_vector_type(8))) float v8f;

#define BATCH   1024
#define MAX_LEN 200
#define MAX_K   8
#define IN_U    256
#define OUT_U   256
#define NEG_PAD (-65535.0f)

// ---------------------------------------------------------------------------
// init: Bm = B_matrix, Bdelta = 0   (MAX_K*MAX_LEN = 1600 floats each)
// ---------------------------------------------------------------------------
__global__ void routing_init_kernel(const float* __restrict__ Bmat,
                                    float* __restrict__ Bm,
                                    float* __restrict__ Bdelta) {
    int i = blockIdx.x * blockDim.x + threadIdx.x;
    if (i < MAX_K * MAX_LEN) {
        Bm[i]     = Bmat[i];
        Bdelta[i] = 0.0f;
    }
}

// ---------------------------------------------------------------------------
// Bm += Bdelta; Bdelta = 0
// ---------------------------------------------------------------------------
__global__ void routing_update_kernel(float* __restrict__ Bm,
                                      float* __restrict__ Bdelta) {
    int i = blockIdx.x * blockDim.x + threadIdx.x;
    if (i < MAX_K * MAX_LEN) {
        Bm[i]    += Bdelta[i];
        Bdelta[i] = 0.0f;
    }
}

// ---------------------------------------------------------------------------
// Projection: low_new[M=204800, 256] = low_capsule[M, 256] @ S[256, 256]
// One wave computes a 16x64 output strip (4 n-tiles), reusing the A fragment.
// M/16 = 12800 row-tiles * 4 strip-columns = 51200 wave tasks, 8 waves/block.
// fp32 WMMA 16x16x4.
// ---------------------------------------------------------------------------
__global__ __launch_bounds__(256)
void proj_kernel(const float* __restrict__ A,   // [204800, 256]
                 const float* __restrict__ S,   // [256, 256]
                 float* __restrict__ C) {       // [204800, 256]
    const int lane = threadIdx.x & 31;
    const int wave = threadIdx.x >> 5;
    const int task = blockIdx.x * 8 + wave;       // 0 .. 51199
    const int stripCol = task & 3;                // 4 strips of 64 cols
    const int tileM    = task >> 2;               // 0 .. 12799
    const int rowBase  = tileM * 16;
    const int nBase    = stripCol * 64;

    const int m  = lane & 15;                     // matrix row (A) / col (B,C)
    const int kp = (lane >> 4) * 2;               // k-pair select

    const float* __restrict__ Arow = A + (size_t)(rowBase + m) * IN_U;

    v8f acc0 = {}, acc1 = {}, acc2 = {}, acc3 = {};
    for (int k0 = 0; k0 < IN_U; k0 += 4) {
        v2f a = *(const v2f*)(Arow + k0 + kp);    // A[m][k0+kp .. +1]
        const float* __restrict__ Sr0 = S + (size_t)(k0 + kp) * OUT_U + nBase + m;
        const float* __restrict__ Sr1 = Sr0 + OUT_U;
        v2f b0; b0.x = Sr0[0];  b0.y = Sr1[0];
        v2f b1; b1.x = Sr0[16]; b1.y = Sr1[16];
        v2f b2; b2.x = Sr0[32]; b2.y = Sr1[32];
        v2f b3; b3.x = Sr0[48]; b3.y = Sr1[48];
        acc0 = __builtin_amdgcn_wmma_f32_16x16x4_f32(false, a, false, b0, (short)0, acc0, false, false);
        acc1 = __builtin_amdgcn_wmma_f32_16x16x4_f32(false, a, false, b1, (short)0, acc1, false, false);
        acc2 = __builtin_amdgcn_wmma_f32_16x16x4_f32(false, a, false, b2, (short)0, acc2, false, false);
        acc3 = __builtin_amdgcn_wmma_f32_16x16x4_f32(false, a, false, b3, (short)0, acc3, false, false);
    }

    const int rowOff = (lane >> 4) * 8;           // C/D: lanes 16-31 hold M+8
    float* __restrict__ Cb = C + (size_t)(rowBase + rowOff) * OUT_U + nBase + m;
#pragma unroll
    for (int r = 0; r < 8; ++r) {
        float* __restrict__ Cr = Cb + (size_t)r * OUT_U;
        Cr[0]  = acc0[r];
        Cr[16] = acc1[r];
        Cr[32] = acc2[r];
        Cr[48] = acc3[r];
    }
}

// ---------------------------------------------------------------------------
// One routing iteration, one block (256 thr = 8 waves) per batch element:
//   W     = softmax_L(mask ? Bm : NEG_PAD)          [8, 200]  (wave w = row w)
//   highR = W @ low_new[b]                          [8, 256]  (WMMA, M padded to 16)
//   high  = squash_K(highR)                          -> d_out on last iteration
//   Bdelta += high @ low_new[b]^T                   [8, 200]  (WMMA, atomics)
// ---------------------------------------------------------------------------
__global__ __launch_bounds__(256)
void routing_iter_kernel(const float* __restrict__ low_new, // [1024,200,256]
                         const float* __restrict__ Bm,      // [8,200]
                         const int*   __restrict__ seq_len, // [1024]
                         float* __restrict__ Bdelta,        // [8,200]
                         float* __restrict__ high_out,      // [1024,8,256]
                         int write_out, int accum_delta) {
    __shared__ float ldsW[16 * MAX_LEN];   // W padded to 16 rows (rows 8..15 zero)
    __shared__ float ldsH[16 * OUT_U];     // high padded to 16 rows

    const int b    = blockIdx.x;
    const int tid  = threadIdx.x;
    const int lane = tid & 31;
    const int wave = tid >> 5;
    const int sl   = seq_len[b];
    const float* __restrict__ ln = low_new + (size_t)b * MAX_LEN * OUT_U;

    // zero pad rows 8..15 of W
    for (int i = tid; i < 8 * MAX_LEN; i += 256) ldsW[8 * MAX_LEN + i] = 0.0f;

    // --- masked softmax over L: wave `wave` owns row k = wave ------------
    {
        const int k = wave;
        const float* __restrict__ Brow = Bm + k * MAX_LEN;
        float mx = -3.4e38f;
        for (int l = lane; l < MAX_LEN; l += 32) {
            float v = (l < sl) ? Brow[l] : NEG_PAD;
            mx = fmaxf(mx, v);
        }
        for (int off = 16; off > 0; off >>= 1) mx = fmaxf(mx, __shfl_xor(mx, off, 32));
        float sum = 0.0f;
        for (int l = lane; l < MAX_LEN; l += 32) {
            float v = (l < sl) ? Brow[l] : NEG_PAD;
            float e = __expf(v - mx);
            ldsW[k * MAX_LEN + l] = e;
            sum += e;
        }
        for (int off = 16; off > 0; off >>= 1) sum += __shfl_xor(sum, off, 32);
        const float inv = 1.0f / sum;
        for (int l = lane; l < MAX_LEN; l += 32) ldsW[k * MAX_LEN + l] *= inv;
    }
    __syncthreads();

    const int m  = lane & 15;
    const int kp = (lane >> 4) * 2;

    // --- highR = W[16,200] @ ln[200,256]; wave owns 32 output cols -------
    {
        const int nBase0 = wave * 32;
        const int nBase1 = nBase0 + 16;
        v8f acc0 = {}, acc1 = {};
        for (int k0 = 0; k0 < MAX_LEN; k0 += 4) {
            v2f a = *(const v2f*)(ldsW + m * MAX_LEN + k0 + kp);   // W[m][k0+kp..+1]
            const float* __restrict__ r0 = ln + (size_t)(k0 + kp) * OUT_U;
            const float* __restrict__ r1 = r0 + OUT_U;
            v2f b0; b0.x = r0[nBase0 + m]; b0.y = r1[nBase0 + m];
            v2f b1; b1.x = r0[nBase1 + m]; b1.y = r1[nBase1 + m];
            acc0 = __builtin_amdgcn_wmma_f32_16x16x4_f32(false, a, false, b0, (short)0, acc0, false, false);
            acc1 = __builtin_amdgcn_wmma_f32_16x16x4_f32(false, a, false, b1, (short)0, acc1, false, false);
        }
        const int rowOff = (lane >> 4) * 8;  // rows 8..15 are zeros (padded A) -> needed zeros in ldsH
#pragma unroll
        for (int r = 0; r < 8; ++r) {
            ldsH[(rowOff + r) * OUT_U + nBase0 + m] = acc0[r];
            ldsH[(rowOff + r) * OUT_U + nBase1 + m] = acc1[r];
        }
    }
    __syncthreads();

    // --- squash over K per output unit o = tid ---------------------------
    {
        const int o = tid;
        float nrm = 0.0f;
#pragma unroll
        for (int k = 0; k < MAX_K; ++k) { float v = ldsH[k * OUT_U + o]; nrm += v * v; }
        const float scale = (nrm / (1.0f + nrm)) * rsqrtf(nrm + 1e-9f);
#pragma unroll
        for (int k = 0; k < MAX_K; ++k) {
            float h = ldsH[k * OUT_U + o] * scale;
            ldsH[k * OUT_U + o] = h;
            if (write_out) high_out[((size_t)b * MAX_K + k) * OUT_U + o] = h;
        }
    }
    __syncthreads();

    // --- Bdelta += high[16,256] @ ln^T[256,200] --------------------------
    if (accum_delta) {
        for (int t = wave; t < 13; t += 8) {          // 13 l-tiles of 16 (200 padded to 208)
            const int lBase = t * 16;
            const int lB    = lBase + m;
            const bool okB  = (lB < MAX_LEN);
            const float* __restrict__ lrow = ln + (size_t)(okB ? lB : 0) * OUT_U;
            v8f acc = {};
            for (int k0 = 0; k0 < OUT_U; k0 += 4) {
                v2f a = *(const v2f*)(ldsH + m * OUT_U + k0 + kp);  // high[m][k0+kp..+1]
                v2f bb;
                bb.x = okB ? lrow[k0 + kp]     : 0.0f;              // ln^T[k][l] = ln[l][k]
                bb.y = okB ? lrow[k0 + kp + 1] : 0.0f;
                acc = __builtin_amdgcn_wmma_f32_16x16x4_f32(false, a, false, bb, (short)0, acc, false, false);
            }
            if (lane < 16) {                    // rows 0..7 hold k=0..7; rows 8..15 are zero
                const int l = lBase + lane;
                if (l < MAX_LEN) {
#pragma unroll
                    for (int r = 0; r < MAX_K; ++r)
                        atomicAdd(&Bdelta[r * MAX_LEN + l], acc[r]);
                }
            }
        }
    }
}

// ---------------------------------------------------------------------------
extern "C" void kernel_launch(void* const* d_in, const int* in_sizes, int n_in,
                              void* d_out, int out_size, void* d_ws, size_t ws_size,
                              hipStream_t stream) {
    const float* low  = (const float*)d_in[0];   // [1024,200,256]
    const float* S    = (const float*)d_in[1];   // [256,256]
    const float* Bmat = (const float*)d_in[2];   // [1,8,200]
    const int*   seq  = (const int*)d_in[3];     // [1024,1]
    float* out = (float*)d_out;                  // [1024,8,256]

    float* low_new = (float*)d_ws;                                   // 209,715,200 B
    float* Bm      = low_new + (size_t)BATCH * MAX_LEN * OUT_U;      // 1600 floats
    float* Bdelta  = Bm + MAX_K * MAX_LEN;                           // 1600 floats

    routing_init_kernel<<<(MAX_K * MAX_LEN + 255) / 256, 256, 0, stream>>>(Bmat, Bm, Bdelta);

    // 204800/16 row-tiles * 4 strips = 51200 wave tasks / 8 waves per block
    proj_kernel<<<51200 / 8, 256, 0, stream>>>(low, S, low_new);

    for (int it = 0; it < 3; ++it) {
        const int last = (it == 2);
        routing_iter_kernel<<<BATCH, 256, 0, stream>>>(low_new, Bm, seq, Bdelta, out,
                                                       last, !last);
        if (!last)
            routing_update_kernel<<<(MAX_K * MAX_LEN + 255) / 256, 256, 0, stream>>>(Bm, Bdelta);
    }
}